// BinGATConv_46273977647979
// MI455X (gfx1250) — compile-verified
//
#include <hip/hip_runtime.h>
#include <hip/hip_bf16.h>

typedef float v2f __attribute__((ext_vector_type(2)));
typedef float v8f __attribute__((ext_vector_type(8)));

#define NN 40000
#define EE 640000
#define GG 64
#define ETOT (EE + NN)

// ---------------------------------------------------------------------------
// fp32 WMMA GEMM:  C[M x OUT] = A[M x K] * W[OUT x K]^T
// One wave (32 lanes) computes a 16x16 tile of C using V_WMMA_F32_16X16X4_F32.
// blockDim = (32, OUT/16); gridDim.x = M/16.
// A fragment (16x4 f32): lane l holds A[l&15, kb] in v0, A[l&15, kb+1] in v1,
//   kb = (l>>4)*2  -> contiguous float2 load from row of A.
// B fragment (4x16 f32): lane l holds B[kb, l&15], B[kb+1, l&15]; since
//   B[k,n] = W[n,k], this is a contiguous float2 load from row n of W.
// D (16x16 f32): lane l, vgpr r -> element (  (l>>4)*8 + r,  l&15 ).
// ---------------------------------------------------------------------------
__global__ __launch_bounds__(256) void wmma_gemm_awt(
    const float* __restrict__ A, const float* __restrict__ W,
    float* __restrict__ C, int K, int OUT)
{
  const int lane  = threadIdx.x;            // 0..31
  const int ntile = threadIdx.y;
  const int rowb  = blockIdx.x * 16;
  const int kb    = (lane >> 4) * 2;
  const int msub  = lane & 15;
  const int ncol  = ntile * 16 + (lane & 15);

  const float* arow = A + (size_t)(rowb + msub) * K;
  const float* wrow = W + (size_t)ncol * K;

  v8f c = {};
  for (int k = 0; k < K; k += 4) {
    v2f a = *(const v2f*)(arow + k + kb);
    v2f b = *(const v2f*)(wrow + k + kb);
    c = __builtin_amdgcn_wmma_f32_16x16x4_f32(
        /*neg_a=*/false, a, /*neg_b=*/false, b,
        /*c_mod=*/(short)0, c, /*reuse_a=*/false, /*reuse_b=*/false);
  }

  const int m0 = rowb + (lane >> 4) * 8;
#pragma unroll
  for (int r = 0; r < 8; ++r)
    C[(size_t)(m0 + r) * OUT + ncol] = c[r];
}

// ---------------------------------------------------------------------------
// per-node attention scores: s = h . a_src ; d = h . a_dst
// ---------------------------------------------------------------------------
__global__ void score_kernel(const float* __restrict__ h,
                             const float* __restrict__ a_src,
                             const float* __restrict__ a_dst,
                             float* __restrict__ s, float* __restrict__ d,
                             int F)
{
  int n = blockIdx.x * blockDim.x + threadIdx.x;
  if (n >= NN) return;
  const float* hp = h + (size_t)n * F;
  float ss = 0.f, dd = 0.f;
  for (int f = 0; f < F; ++f) {
    float v = hp[f];
    ss += v * a_src[f];
    dd += v * a_dst[f];
  }
  s[n] = ss;
  d[n] = dd;
}

// order-preserving float<->uint mapping for atomic max on signed floats
__device__ __forceinline__ unsigned f2ord(float f) {
  unsigned u = __float_as_uint(f);
  return (u & 0x80000000u) ? ~u : (u | 0x80000000u);
}
__device__ __forceinline__ float ord2f(unsigned u) {
  unsigned b = (u & 0x80000000u) ? (u ^ 0x80000000u) : ~u;
  return __uint_as_float(b);
}

__device__ __forceinline__ void edge_sd(const int* ei, int t, int& src, int& dst) {
  if (t < EE) { src = ei[t]; dst = ei[EE + t]; }
  else        { src = t - EE; dst = t - EE; }      // self loops
}

// pass 1: e = leaky_relu(s[src]+d[dst], 0.2); segment max over dst
__global__ void edge_score_max(const int* __restrict__ ei,
                               const float* __restrict__ s,
                               const float* __restrict__ d,
                               float* __restrict__ ebuf,
                               unsigned* __restrict__ m_u)
{
  int t = blockIdx.x * blockDim.x + threadIdx.x;
  if (t >= ETOT) return;
  int src, dst; edge_sd(ei, t, src, dst);
  float e = s[src] + d[dst];
  e = (e > 0.f) ? e : 0.2f * e;
  ebuf[t] = e;
  atomicMax(m_u + dst, f2ord(e));
}

// pass 2: ex = exp(e - m[dst]); segment sum over dst
__global__ void edge_exp_sum(const int* __restrict__ ei,
                             float* __restrict__ ebuf,
                             const unsigned* __restrict__ m_u,
                             float* __restrict__ z)
{
  int t = blockIdx.x * blockDim.x + threadIdx.x;
  if (t >= ETOT) return;
  int src, dst; edge_sd(ei, t, src, dst);
  float ex = expf(ebuf[t] - ord2f(m_u[dst]));
  ebuf[t] = ex;
  atomicAdd(z + dst, ex);
}

// pass 3: agg[dst,:] += (ex/z[dst]) * h[src,:]   (one block / edge, F threads)
__global__ void edge_aggregate(const int* __restrict__ ei,
                               const float* __restrict__ ebuf,
                               const float* __restrict__ z,
                               const float* __restrict__ h,
                               float* __restrict__ agg, int F)
{
  int t = blockIdx.x;
  int src, dst; edge_sd(ei, t, src, dst);
  float alpha = ebuf[t] / z[dst];
  int f = threadIdx.x;
  atomicAdd(&agg[(size_t)dst * F + f], alpha * h[(size_t)src * F + f]);
}

// bias + relu + layernorm, in place. one block / node, F threads.
__global__ void post_relu_ln(float* __restrict__ agg,
                             const float* __restrict__ bias,
                             const float* __restrict__ gamma,
                             const float* __restrict__ beta, int F)
{
  __shared__ float red[128];
  int n = blockIdx.x, f = threadIdx.x;
  float v = agg[(size_t)n * F + f] + bias[f];
  v = v > 0.f ? v : 0.f;
  red[f] = v;
  __syncthreads();
  for (int st = F >> 1; st > 0; st >>= 1) {
    if (f < st) red[f] += red[f + st];
    __syncthreads();
  }
  float mu = red[0] / (float)F;
  __syncthreads();
  float dv = v - mu;
  red[f] = dv * dv;
  __syncthreads();
  for (int st = F >> 1; st > 0; st >>= 1) {
    if (f < st) red[f] += red[f + st];
    __syncthreads();
  }
  float var = red[0] / (float)F;
  agg[(size_t)n * F + f] = gamma[f] * dv * rsqrtf(var + 1e-5f) + beta[f];
}

// global mean pool accumulation: one block / node, 64 threads
__global__ void pool_kernel(const float* __restrict__ h,
                            const int* __restrict__ batch,
                            float* __restrict__ sums, float* __restrict__ cnt)
{
  int n = blockIdx.x, f = threadIdx.x;
  int g = batch[n];
  atomicAdd(&sums[(size_t)g * 64 + f], h[(size_t)n * 64 + f]);
  if (f == 0) atomicAdd(&cnt[g], 1.0f);
}

// pooled = sums/max(cnt,1); out = (pooled @ Wl^T + bl) @ Wc^T + bc
__global__ void head_kernel(const float* __restrict__ sums,
                            const float* __restrict__ cnt,
                            const float* __restrict__ Wl, const float* __restrict__ bl,
                            const float* __restrict__ Wc, const float* __restrict__ bc,
                            float* __restrict__ out)
{
  __shared__ float pooled[64];
  __shared__ float lin[64];
  int g = blockIdx.x, o = threadIdx.x;
  float c = cnt[g]; c = c > 1.f ? c : 1.f;
  pooled[o] = sums[(size_t)g * 64 + o] / c;
  __syncthreads();
  float acc = bl[o];
  for (int k = 0; k < 64; ++k) acc += pooled[k] * Wl[o * 64 + k];
  lin[o] = acc;
  __syncthreads();
  if (o == 0) {
    float r = bc[0];
    for (int k = 0; k < 64; ++k) r += lin[k] * Wc[k];
    out[g] = r;
  }
}

extern "C" void kernel_launch(void* const* d_in, const int* in_sizes, int n_in,
                              void* d_out, int out_size, void* d_ws, size_t ws_size,
                              hipStream_t stream) {
  const float* x      = (const float*)d_in[0];
  const int*   ei     = (const int*)  d_in[1];
  const int*   batch  = (const int*)  d_in[2];
  const float* W1     = (const float*)d_in[3];
  const float* a1s    = (const float*)d_in[4];
  const float* a1d    = (const float*)d_in[5];
  const float* b1     = (const float*)d_in[6];
  const float* g1     = (const float*)d_in[7];
  const float* be1    = (const float*)d_in[8];
  const float* W2     = (const float*)d_in[9];
  const float* a2s    = (const float*)d_in[10];
  const float* a2d    = (const float*)d_in[11];
  const float* b2     = (const float*)d_in[12];
  const float* g2     = (const float*)d_in[13];
  const float* be2    = (const float*)d_in[14];
  const float* Wl     = (const float*)d_in[15];
  const float* bl     = (const float*)d_in[16];
  const float* Wc     = (const float*)d_in[17];
  const float* bc     = (const float*)d_in[18];
  float* out = (float*)d_out;

  // workspace layout (floats)
  float* ws   = (float*)d_ws;
  float* h1   = ws;                      // N*128 (reused: h2 = h1, agg2 = h1+N*64)
  float* agg1 = h1 + (size_t)NN * 128;   // N*128 (in-place LN output = layer-2 input)
  float* s    = agg1 + (size_t)NN * 128; // N
  float* dsc  = s + NN;                  // N
  unsigned* m_u = (unsigned*)(dsc + NN); // N
  float* z    = (float*)(m_u + NN);      // N
  float* ebuf = z + NN;                  // ETOT
  float* sums = ebuf + ETOT;             // G*64
  float* cnt  = sums + (size_t)GG * 64;  // G
  float* h2   = h1;                      // N*64
  float* agg2 = h1 + (size_t)NN * 64;    // N*64

  const int EB = (ETOT + 255) / 256;

  // ---------------- layer 1 (F = 128) ----------------
  hipMemsetAsync(agg1, 0, (size_t)NN * 128 * sizeof(float), stream);
  hipMemsetAsync(m_u,  0, (size_t)NN * sizeof(unsigned), stream);
  hipMemsetAsync(z,    0, (size_t)NN * sizeof(float), stream);

  wmma_gemm_awt<<<NN / 16, dim3(32, 8), 0, stream>>>(x, W1, h1, 128, 128);
  score_kernel<<<(NN + 255) / 256, 256, 0, stream>>>(h1, a1s, a1d, s, dsc, 128);
  edge_score_max<<<EB, 256, 0, stream>>>(ei, s, dsc, ebuf, m_u);
  edge_exp_sum  <<<EB, 256, 0, stream>>>(ei, ebuf, m_u, z);
  edge_aggregate<<<ETOT, 128, 0, stream>>>(ei, ebuf, z, h1, agg1, 128);
  post_relu_ln  <<<NN, 128, 0, stream>>>(agg1, b1, g1, be1, 128);

  // ---------------- layer 2 (F = 64) -----------------
  hipMemsetAsync(agg2, 0, (size_t)NN * 64 * sizeof(float), stream);
  hipMemsetAsync(m_u,  0, (size_t)NN * sizeof(unsigned), stream);
  hipMemsetAsync(z,    0, (size_t)NN * sizeof(float), stream);

  wmma_gemm_awt<<<NN / 16, dim3(32, 4), 0, stream>>>(agg1, W2, h2, 128, 64);
  score_kernel<<<(NN + 255) / 256, 256, 0, stream>>>(h2, a2s, a2d, s, dsc, 64);
  edge_score_max<<<EB, 256, 0, stream>>>(ei, s, dsc, ebuf, m_u);
  edge_exp_sum  <<<EB, 256, 0, stream>>>(ei, ebuf, m_u, z);
  edge_aggregate<<<ETOT, 64, 0, stream>>>(ei, ebuf, z, h2, agg2, 64);
  post_relu_ln  <<<NN, 64, 0, stream>>>(agg2, b2, g2, be2, 64);

  // ---------------- pool + head ----------------------
  hipMemsetAsync(sums, 0, (size_t)(GG * 64 + GG) * sizeof(float), stream);
  pool_kernel<<<NN, 64, 0, stream>>>(agg2, batch, sums, cnt);
  head_kernel<<<GG, 64, 0, stream>>>(sums, cnt, Wl, bl, Wc, bc, out);
}